// KGAT_89713276879535
// MI455X (gfx1250) — compile-verified
//
#include <hip/hip_runtime.h>

// ---------------------------------------------------------------------------
// KGAT bi-interaction on MI455X (gfx1250, wave32).
//   - wave-per-edge scatter: coalesced 128B gathers + per-lane f32 L2 atomics
//   - 16-node tiles through V_WMMA_F32_16X16X4_F32 (fp32 matrix cores)
//   - weights staged in LDS, biases from L0/L2
//   - l2norm via half-wave shfl_xor reduction (wave32)
// ---------------------------------------------------------------------------

typedef __attribute__((ext_vector_type(2))) float v2f;
typedef __attribute__((ext_vector_type(8))) float v8f;

#define N_USERS 50000
#define N_TOT   150000
#define OUT_STRIDE 176   // 64 + 64 + 32 + 16

__device__ __forceinline__ float leaky01(float x) {
    return x > 0.0f ? x : 0.01f * x;
}

// ---------------------------------------------------------------------------
// Init: ego = concat(user, entity); out[:, 0:64] = ego (unnormalized, outs[0])
// ---------------------------------------------------------------------------
__global__ __launch_bounds__(256)
void kgat_init_kernel(const float* __restrict__ user,
                      const float* __restrict__ ent,
                      float* __restrict__ ego,
                      float* __restrict__ out)
{
    int i = blockIdx.x * blockDim.x + threadIdx.x;   // over N_TOT*64
    if (i >= N_TOT * 64) return;
    int node = i >> 6;
    int d    = i & 63;
    float v = (node < N_USERS) ? user[(size_t)node * 64 + d]
                               : ent[(size_t)(node - N_USERS) * 64 + d];
    ego[i] = v;
    out[(size_t)node * OUT_STRIDE + d] = v;
}

__global__ __launch_bounds__(256)
void kgat_zero_kernel(float* __restrict__ p, int n)
{
    int i = blockIdx.x * blockDim.x + threadIdx.x;
    if (i < n) p[i] = 0.0f;
}

// ---------------------------------------------------------------------------
// Scatter: side[h] += A[e] * ego[t].  One wave (32 lanes) per edge.
// DIN=64 -> 2 coalesced 128B gather rounds; DIN=32 -> 1.
// Atomics land in L2 (tables are L2-resident: 38MB << 192MB).
// ---------------------------------------------------------------------------
template<int DIN>
__global__ __launch_bounds__(256)
void kgat_scatter_kernel(const float* __restrict__ ego,
                         const float* __restrict__ Avals,
                         const int*   __restrict__ eh,
                         const int*   __restrict__ et,
                         float* __restrict__ side,
                         int E)
{
    int wave = (blockIdx.x * blockDim.x + threadIdx.x) >> 5;
    int lane = threadIdx.x & 31;
    if (wave >= E) return;
    int   h = eh[wave];
    int   t = et[wave];
    float a = Avals[wave];
    const float* src = ego  + (size_t)t * DIN;
    float*       dst = side + (size_t)h * DIN;
#pragma unroll
    for (int c = 0; c < DIN; c += 32) {
        float v = a * src[c + lane];
        atomicAdd(&dst[c + lane], v);
    }
}

// ---------------------------------------------------------------------------
// Layer: one wave handles a 16-node tile.
//   sum_emb = leaky((ego+side) @ Wg + bg)
//   bi_emb  = leaky((ego*side) @ Wb + bb)
//   ego'    = sum_emb + bi_emb ; out = l2norm(ego')
// WMMA f32 16x16x4: A frag = float2 per lane (lane = kh*16 + m),
// C/D: vgpr r, lane -> (row r + 8*(lane>>4), col lane&15).
// ---------------------------------------------------------------------------
template<int DIN, int DOUT>
__global__ __launch_bounds__(256)
void kgat_layer_kernel(const float* __restrict__ ego,
                       const float* __restrict__ side,
                       const float* __restrict__ Wg,
                       const float* __restrict__ bg,
                       const float* __restrict__ Wb,
                       const float* __restrict__ bb,
                       float* __restrict__ ego_next,
                       float* __restrict__ out,
                       int col_off)
{
    __shared__ float sWg[DIN * DOUT];
    __shared__ float sWb[DIN * DOUT];
    for (int i = threadIdx.x; i < DIN * DOUT; i += 256) {
        sWg[i] = Wg[i];
        sWb[i] = Wb[i];
    }
    __syncthreads();

    const int lane  = threadIdx.x & 31;
    const int tile  = blockIdx.x * 8 + (threadIdx.x >> 5);
    const int node0 = tile * 16;
    if (node0 >= N_TOT) return;

    const int m  = lane & 15;   // A-frag row / B-frag col / D col
    const int kh = lane >> 4;   // K-half select

    constexpr int NT = DOUT / 16;
    v8f accG[NT];
    v8f accB[NT];
    v8f zero = {};
#pragma unroll
    for (int nt = 0; nt < NT; ++nt) { accG[nt] = zero; accB[nt] = zero; }

    const float* egoRow  = ego  + (size_t)(node0 + m) * DIN;
    const float* sideRow = side + (size_t)(node0 + m) * DIN;

#pragma unroll
    for (int k0 = 0; k0 < DIN; k0 += 4) {
        const int kk = k0 + 2 * kh;
        float e0 = egoRow[kk],  e1 = egoRow[kk + 1];
        float s0 = sideRow[kk], s1 = sideRow[kk + 1];
        v2f aS; aS.x = e0 + s0; aS.y = e1 + s1;   // (ego + side) fragment
        v2f aB; aB.x = e0 * s0; aB.y = e1 * s1;   // (ego * side) fragment
#pragma unroll
        for (int nt = 0; nt < NT; ++nt) {
            const int nc = nt * 16 + m;
            v2f bG; bG.x = sWg[kk * DOUT + nc]; bG.y = sWg[(kk + 1) * DOUT + nc];
            v2f bB; bB.x = sWb[kk * DOUT + nc]; bB.y = sWb[(kk + 1) * DOUT + nc];
            accG[nt] = __builtin_amdgcn_wmma_f32_16x16x4_f32(
                false, aS, false, bG, (short)0, accG[nt], false, false);
            accB[nt] = __builtin_amdgcn_wmma_f32_16x16x4_f32(
                false, aB, false, bB, (short)0, accB[nt], false, false);
        }
    }

    // Epilogue: bias + leaky + sum, then per-row l2 norm (16-lane halves)
#pragma unroll
    for (int r = 0; r < 8; ++r) {
        const int mr = r + 8 * kh;            // output row within tile
        float vals[NT];
        float ssq = 0.0f;
#pragma unroll
        for (int nt = 0; nt < NT; ++nt) {
            const int nc = nt * 16 + m;
            float vg = leaky01(accG[nt][r] + bg[nc]);
            float vb = leaky01(accB[nt][r] + bb[nc]);
            float v  = vg + vb;
            vals[nt] = v;
            ssq += v * v;
        }
        // reduce over the 16 lanes of this half-wave (xor stays inside half)
        ssq += __shfl_xor(ssq, 8, 32);
        ssq += __shfl_xor(ssq, 4, 32);
        ssq += __shfl_xor(ssq, 2, 32);
        ssq += __shfl_xor(ssq, 1, 32);
        float nrm = sqrtf(ssq);
        float inv = 1.0f / fmaxf(nrm, 1e-12f);

        const size_t row = (size_t)(node0 + mr);
#pragma unroll
        for (int nt = 0; nt < NT; ++nt) {
            const int nc = nt * 16 + m;
            ego_next[row * DOUT + nc]            = vals[nt];
            out[row * OUT_STRIDE + col_off + nc] = vals[nt] * inv;
        }
    }
}

// ---------------------------------------------------------------------------
extern "C" void kernel_launch(void* const* d_in, const int* in_sizes, int n_in,
                              void* d_out, int out_size, void* d_ws, size_t ws_size,
                              hipStream_t stream)
{
    const float* user  = (const float*)d_in[0];
    const float* ent   = (const float*)d_in[1];
    const float* Avals = (const float*)d_in[2];
    const int*   eh    = (const int*)d_in[3];
    const int*   et    = (const int*)d_in[4];
    const float* Wg0 = (const float*)d_in[5];
    const float* bg0 = (const float*)d_in[6];
    const float* Wb0 = (const float*)d_in[7];
    const float* bb0 = (const float*)d_in[8];
    const float* Wg1 = (const float*)d_in[9];
    const float* bg1 = (const float*)d_in[10];
    const float* Wb1 = (const float*)d_in[11];
    const float* bb1 = (const float*)d_in[12];
    const float* Wg2 = (const float*)d_in[13];
    const float* bg2 = (const float*)d_in[14];
    const float* Wb2 = (const float*)d_in[15];
    const float* bb2 = (const float*)d_in[16];

    float* out   = (float*)d_out;
    float* ego_a = (float*)d_ws;
    float* ego_b = ego_a + (size_t)N_TOT * 64;
    float* sideb = ego_b + (size_t)N_TOT * 64;

    const int E = in_sizes[2];   // 4,000,000

    const int tiles       = N_TOT / 16;              // 9375
    const int layerBlocks = (tiles + 7) / 8;
    const int scatBlocks  = (E + 7) / 8;             // 8 waves/block, wave/edge

    // init: ego_a <- concat embeddings; out[:,0:64] = ego
    kgat_init_kernel<<<(N_TOT * 64 + 255) / 256, 256, 0, stream>>>(user, ent, ego_a, out);

    // ---- layer 0: 64 -> 64 ----
    kgat_zero_kernel<<<(N_TOT * 64 + 255) / 256, 256, 0, stream>>>(sideb, N_TOT * 64);
    kgat_scatter_kernel<64><<<scatBlocks, 256, 0, stream>>>(ego_a, Avals, eh, et, sideb, E);
    kgat_layer_kernel<64, 64><<<layerBlocks, 256, 0, stream>>>(
        ego_a, sideb, Wg0, bg0, Wb0, bb0, ego_b, out, 64);

    // ---- layer 1: 64 -> 32 ----
    kgat_zero_kernel<<<(N_TOT * 64 + 255) / 256, 256, 0, stream>>>(sideb, N_TOT * 64);
    kgat_scatter_kernel<64><<<scatBlocks, 256, 0, stream>>>(ego_b, Avals, eh, et, sideb, E);
    kgat_layer_kernel<64, 32><<<layerBlocks, 256, 0, stream>>>(
        ego_b, sideb, Wg1, bg1, Wb1, bb1, ego_a, out, 128);

    // ---- layer 2: 32 -> 16 ----
    kgat_zero_kernel<<<(N_TOT * 32 + 255) / 256, 256, 0, stream>>>(sideb, N_TOT * 32);
    kgat_scatter_kernel<32><<<scatBlocks, 256, 0, stream>>>(ego_a, Avals, eh, et, sideb, E);
    kgat_layer_kernel<32, 16><<<layerBlocks, 256, 0, stream>>>(
        ego_a, sideb, Wg2, bg2, Wb2, bb2, ego_b, out, 160);
}